// SoftTopKExtractor_36335423324463
// MI455X (gfx1250) — compile-verified
//
#include <hip/hip_runtime.h>
#include <stdint.h>

// ---------------------------------------------------------------------------
// SoftTopKExtractor for MI455X (gfx1250).
// Bandwidth-bound problem (64 MB in, ~1 KB out, ~few compares/elem). No GEMM
// => WMMA not applicable. CDNA5-specific path used: async global->LDS loads
// (global_load_async_to_lds_b128, ASYNCcnt) for the 9x9 local-max tile
// staging + s_wait_asynccnt.
// Exact quantile via 3-pass radix-select (11+11+10 bits, 2048/2048/1024 bins;
// data fits in 192 MB L2, so passes 2..3 stream from L2 at >HBM bandwidth).
// Total data sweeps: 3 (select) + 1 (peaks) = 4 x 64 MB.
// ---------------------------------------------------------------------------

#define NB_B   16
#define IMG_W  1024
#define IMG_H  1024
#define NPIX   (IMG_W * IMG_H)
#define KTHR   104857u          // int((1-0.9)*1048576) -> k-th largest (1-based)
#define TOPK   5
#define HBLK   128              // hist blocks per image
#define PBLK   128              // peak blocks per image (8 output rows each)
#define MAXBINS 2048
#define NEG_INF (-__builtin_inff())

struct WS {
  unsigned hist[NB_B][MAXBINS];
  unsigned sel_prefix[NB_B];
  unsigned sel_K[NB_B];
  float    thr[NB_B];
  float    fmax_part[NB_B][HBLK];
  unsigned fidx_part[NB_B][HBLK];
  float    pval[NB_B][PBLK][TOPK];
  unsigned pidx[NB_B][PBLK][TOPK];
};

// Monotonic float<->uint mapping (ascending uint order == ascending float order)
__device__ __forceinline__ unsigned f2mono(float f) {
  unsigned u = __float_as_uint(f);
  return (u & 0x80000000u) ? ~u : (u | 0x80000000u);
}
__device__ __forceinline__ float mono2f(unsigned u) {
  unsigned b = (u & 0x80000000u) ? (u & 0x7fffffffu) : ~u;
  return __uint_as_float(b);
}
// Ordering used by lax.top_k: value desc, index asc on ties.
__device__ __forceinline__ bool better(float v1, unsigned i1, float v2, unsigned i2) {
  return (v1 > v2) || (v1 == v2 && i1 < i2);
}
// Merge two sorted 5-lists into the top-5 (kept in a).
__device__ __forceinline__ void merge5(float* av, unsigned* ai,
                                       const float* bv, const unsigned* bi) {
  float ov[TOPK]; unsigned oi[TOPK];
  int ia = 0, ib = 0;
#pragma unroll
  for (int j = 0; j < TOPK; ++j) {
    bool ta = better(av[ia], ai[ia], bv[ib], bi[ib]);
    ov[j] = ta ? av[ia] : bv[ib];
    oi[j] = ta ? ai[ia] : bi[ib];
    if (ta) ++ia; else ++ib;
  }
#pragma unroll
  for (int j = 0; j < TOPK; ++j) { av[j] = ov[j]; ai[j] = oi[j]; }
}

// ---------------------------------------------------------------------------
__global__ void init_kernel(WS* __restrict__ ws) {
  const int b = blockIdx.x, t = threadIdx.x;
  for (int i = t; i < MAXBINS; i += 256) ws->hist[b][i] = 0u;
  if (b == 0 && t < NB_B) { ws->sel_prefix[t] = 0u; ws->sel_K[t] = KTHR; }
}

// One radix-select counting pass (NBINS bins over the bit-field at SHIFT).
// Pass 0 also produces per-block argmax partials for the no-peak fallback.
template <int SHIFT, int NBINS, bool ARGMAX>
__global__ void __launch_bounds__(256)
hist_kernel(const float* __restrict__ cmap, WS* __restrict__ ws) {
  const int b = blockIdx.y, blk = blockIdx.x, t = threadIdx.x;
  __shared__ unsigned lh[NBINS];
  __shared__ float    sv[256];
  __shared__ unsigned si[256];
  for (int i = t; i < NBINS; i += 256) lh[i] = 0u;
  const unsigned prefix = ws->sel_prefix[b];   // uniform -> scalar load
  __syncthreads();
  constexpr int HB = SHIFT + 11;               // bits consumed incl. this pass
  constexpr unsigned hi_mask =
      (HB >= 32) ? 0u : (unsigned)~((1ull << HB) - 1ull);
  const float4* src = (const float4*)(cmap + (size_t)b * NPIX);
  float vmax = NEG_INF; unsigned vidx = 0u;
#pragma unroll 1
  for (int i = 0; i < 8; ++i) {
    int q = blk * 2048 + t + 256 * i;
    if (i + 1 < 8) __builtin_prefetch((const void*)(src + q + 256), 0, 0);
    float4 v = src[q];
    float e[4] = {v.x, v.y, v.z, v.w};
#pragma unroll
    for (int j = 0; j < 4; ++j) {
      unsigned u = f2mono(e[j]);
      if ((u & hi_mask) == prefix)
        atomicAdd(&lh[(u >> SHIFT) & (unsigned)(NBINS - 1)], 1u);
      if (ARGMAX && e[j] > vmax) { vmax = e[j]; vidx = (unsigned)(q * 4 + j); }
    }
  }
  __syncthreads();
  for (int i = t; i < NBINS; i += 256)
    if (lh[i]) atomicAdd(&ws->hist[b][i], lh[i]);
  if (ARGMAX) {
    sv[t] = vmax; si[t] = vidx;
    __syncthreads();
    for (int s = 128; s >= 1; s >>= 1) {
      if (t < s && better(sv[t + s], si[t + s], sv[t], si[t])) {
        sv[t] = sv[t + s]; si[t] = si[t + s];
      }
      __syncthreads();
    }
    if (t == 0) { ws->fmax_part[b][blk] = sv[0]; ws->fidx_part[b][blk] = si[0]; }
  }
}

// Pick the bin containing descending-rank K, refine state, clear histogram.
template <int SHIFT, int NBINS>
__global__ void scan_kernel(WS* __restrict__ ws) {
  const int b = blockIdx.x, t = threadIdx.x;
  if (t == 0) {
    unsigned K = ws->sel_K[b];
    unsigned above = 0; int sel = 0;
    for (int bin = NBINS - 1; bin >= 0; --bin) {
      unsigned c = ws->hist[b][bin];
      if (above + c >= K) { sel = bin; break; }
      above += c;
    }
    ws->sel_K[b] = K - above;
    unsigned p = ws->sel_prefix[b] | ((unsigned)sel << SHIFT);
    ws->sel_prefix[b] = p;
    if (SHIFT == 0) ws->thr[b] = mono2f(p);   // exact dyn_thr element value
  }
  __syncthreads();
  for (int i = t; i < NBINS; i += 256) ws->hist[b][i] = 0u;  // next pass
}

// 9x9 local-max + peak top-5 per block of 8 output rows.
// 16-row tile (64 KB) staged with global_load_async_to_lds_b128 (ASYNCcnt),
// separable max via a 16-row rowmax plane (64 KB). 128 KB dynamic LDS
// (well inside the 320 KB/WGP budget).
__global__ void __launch_bounds__(256)
peak_kernel(const float* __restrict__ cmap, WS* __restrict__ ws) {
  extern __shared__ float smem[];
  float* tile = smem;                  // [16][1024]
  float* rmax = smem + 16 * IMG_W;     // [16][1024]
  const int b = blockIdx.y, rb = blockIdx.x, t = threadIdx.x;
  const int row0 = rb * 8;
  const float* img = cmap + (size_t)b * NPIX;
  const unsigned voff = (unsigned)(t * 16);           // 16 B per lane
#pragma unroll 1
  for (int lr = 0; lr < 16; ++lr) {
    int grow = row0 - 4 + lr;
    unsigned laddr = (unsigned)(size_t)(tile + lr * IMG_W) + voff;
    if (0 <= grow && grow < IMG_H) {
      unsigned long long gbase =
          (unsigned long long)(size_t)(img + (size_t)grow * IMG_W);
      // CDNA5 async copy: per-lane 16 B global -> LDS, tracked by ASYNCcnt.
      asm volatile("global_load_async_to_lds_b128 %0, %1, %2"
                   :: "v"(laddr), "v"(voff), "s"(gbase) : "memory");
    } else {
      float* dst = tile + lr * IMG_W + t * 4;         // -inf padding rows
      dst[0] = NEG_INF; dst[1] = NEG_INF; dst[2] = NEG_INF; dst[3] = NEG_INF;
    }
  }
  asm volatile("s_wait_asynccnt 0x0" ::: "memory");
  __syncthreads();

  // Phase 1: horizontal 9-tap max (with -inf padding at column borders).
  for (int p = t; p < 16 * IMG_W; p += 256) {
    int c = p & (IMG_W - 1);
    int base = p - c;
    float m = NEG_INF;
#pragma unroll
    for (int d = -4; d <= 4; ++d) {
      int cc = c + d;
      if (0 <= cc && cc < IMG_W) m = fmaxf(m, tile[base + cc]);
    }
    rmax[p] = m;
  }
  __syncthreads();

  // Phase 2: vertical 9-tap max of rowmax, peak test, per-thread top-5.
  float tv[TOPK]; unsigned ti[TOPK];
#pragma unroll
  for (int j = 0; j < TOPK; ++j) { tv[j] = NEG_INF; ti[j] = 0xffffffffu; }
  const float thr = ws->thr[b];
  for (int i = 0; i < 8; ++i) {
    const int lrow = i + 4;
#pragma unroll
    for (int cq = 0; cq < 4; ++cq) {
      const int c = t + 256 * cq;
      const float x = tile[lrow * IMG_W + c];
      float lm = NEG_INF;
#pragma unroll
      for (int d = -4; d <= 4; ++d)
        lm = fmaxf(lm, rmax[(lrow + d) * IMG_W + c]);
      if (x == lm && x > thr) {
        unsigned idx = (unsigned)((row0 + i) * IMG_W + c);
        if (better(x, idx, tv[TOPK - 1], ti[TOPK - 1])) {
          tv[TOPK - 1] = x; ti[TOPK - 1] = idx;
#pragma unroll
          for (int j = TOPK - 1; j > 0; --j) {
            if (better(tv[j], ti[j], tv[j - 1], ti[j - 1])) {
              float fv = tv[j]; tv[j] = tv[j - 1]; tv[j - 1] = fv;
              unsigned fi = ti[j]; ti[j] = ti[j - 1]; ti[j - 1] = fi;
            }
          }
        }
      }
    }
  }
  __syncthreads();                       // done reading tile/rmax
  float*    rv = tile;                   // reuse LDS for reduction scratch
  unsigned* ri = (unsigned*)(tile + 256 * TOPK);
#pragma unroll
  for (int j = 0; j < TOPK; ++j) { rv[t * TOPK + j] = tv[j]; ri[t * TOPK + j] = ti[j]; }
  __syncthreads();
  for (int s = 128; s >= 1; s >>= 1) {
    if (t < s) {
      float a_[TOPK], b_[TOPK]; unsigned ai_[TOPK], bi_[TOPK];
#pragma unroll
      for (int j = 0; j < TOPK; ++j) {
        a_[j] = rv[t * TOPK + j];        ai_[j] = ri[t * TOPK + j];
        b_[j] = rv[(t + s) * TOPK + j];  bi_[j] = ri[(t + s) * TOPK + j];
      }
      merge5(a_, ai_, b_, bi_);
#pragma unroll
      for (int j = 0; j < TOPK; ++j) { rv[t * TOPK + j] = a_[j]; ri[t * TOPK + j] = ai_[j]; }
    }
    __syncthreads();
  }
  if (t == 0) {
#pragma unroll
    for (int j = 0; j < TOPK; ++j) {
      ws->pval[b][rb][j] = rv[j];
      ws->pidx[b][rb][j] = ri[j];
    }
  }
}

// Merge block partials, apply reference's coords/labels logic, emit output.
__global__ void __launch_bounds__(256)
final_kernel(const WS* __restrict__ ws, float* __restrict__ out) {
  const int b = blockIdx.x, t = threadIdx.x;
  __shared__ float    rv[256 * TOPK];
  __shared__ unsigned ri[256 * TOPK];
  __shared__ float    av[256];
  __shared__ unsigned ai[256];
  if (t < PBLK) {
#pragma unroll
    for (int j = 0; j < TOPK; ++j) {
      rv[t * TOPK + j] = ws->pval[b][t][j];
      ri[t * TOPK + j] = ws->pidx[b][t][j];
    }
    av[t] = ws->fmax_part[b][t]; ai[t] = ws->fidx_part[b][t];
  } else {
#pragma unroll
    for (int j = 0; j < TOPK; ++j) { rv[t * TOPK + j] = NEG_INF; ri[t * TOPK + j] = 0xffffffffu; }
    av[t] = NEG_INF; ai[t] = 0xffffffffu;
  }
  __syncthreads();
  for (int s = 128; s >= 1; s >>= 1) {
    if (t < s) {
      float a_[TOPK], b_[TOPK]; unsigned ai_[TOPK], bi_[TOPK];
#pragma unroll
      for (int j = 0; j < TOPK; ++j) {
        a_[j] = rv[t * TOPK + j];        ai_[j] = ri[t * TOPK + j];
        b_[j] = rv[(t + s) * TOPK + j];  bi_[j] = ri[(t + s) * TOPK + j];
      }
      merge5(a_, ai_, b_, bi_);
#pragma unroll
      for (int j = 0; j < TOPK; ++j) { rv[t * TOPK + j] = a_[j]; ri[t * TOPK + j] = ai_[j]; }
      if (better(av[t + s], ai[t + s], av[t], ai[t])) { av[t] = av[t + s]; ai[t] = ai[t + s]; }
    }
    __syncthreads();
  }
  if (t == 0) {
    float topv[TOPK]; unsigned topi[TOPK];
#pragma unroll
    for (int j = 0; j < TOPK; ++j) { topv[j] = rv[j]; topi[j] = ri[j]; }
    unsigned fidx = ai[0];
    bool has_any = topv[0] > NEG_INF;
    float xs[TOPK], ys[TOPK];
#pragma unroll
    for (int j = 0; j < TOPK; ++j) {
      xs[j] = (float)(topi[j] % IMG_W);
      ys[j] = (float)(topi[j] / IMG_W);
    }
    if (!has_any) { xs[0] = (float)(fidx % IMG_W); ys[0] = (float)(fidx / IMG_W); }
    float pm = topv[0];
    int nv = 0;
#pragma unroll
    for (int j = 0; j < TOPK; ++j)
      if (topv[j] >= pm * 0.5f && topv[j] > NEG_INF) ++nv;
    if (nv < 1) nv = 1;
#pragma unroll
    for (int j = 0; j < TOPK; ++j) {
      bool keep = j < nv;
      out[(b * TOPK + j) * 2 + 0] = keep ? xs[j] : -1.0f;   // coords x
      out[(b * TOPK + j) * 2 + 1] = keep ? ys[j] : -1.0f;   // coords y
      out[NB_B * TOPK * 2 + b * TOPK + j] = keep ? 1.0f : -1.0f;  // labels
    }
  }
}

// ---------------------------------------------------------------------------
extern "C" void kernel_launch(void* const* d_in, const int* in_sizes, int n_in,
                              void* d_out, int out_size, void* d_ws, size_t ws_size,
                              hipStream_t stream) {
  (void)in_sizes; (void)n_in; (void)out_size; (void)ws_size;
  const float* cmap = (const float*)d_in[0];
  float* out = (float*)d_out;
  WS* ws = (WS*)d_ws;

  const size_t smem = (size_t)2 * 16 * IMG_W * sizeof(float);   // 128 KB
  (void)hipFuncSetAttribute((const void*)peak_kernel,
                            hipFuncAttributeMaxDynamicSharedMemorySize,
                            (int)smem);

  init_kernel<<<NB_B, 256, 0, stream>>>(ws);
  // Radix select: 11 + 11 + 10 bits -> 3 full-data passes (pass 1 from HBM,
  // passes 2-3 from the 192 MB L2 which holds the whole 64 MB tensor).
  hist_kernel<21, 2048, true ><<<dim3(HBLK, NB_B), 256, 0, stream>>>(cmap, ws);
  scan_kernel<21, 2048><<<NB_B, 256, 0, stream>>>(ws);
  hist_kernel<10, 2048, false><<<dim3(HBLK, NB_B), 256, 0, stream>>>(cmap, ws);
  scan_kernel<10, 2048><<<NB_B, 256, 0, stream>>>(ws);
  hist_kernel< 0, 1024, false><<<dim3(HBLK, NB_B), 256, 0, stream>>>(cmap, ws);
  scan_kernel< 0, 1024><<<NB_B, 256, 0, stream>>>(ws);
  peak_kernel<<<dim3(PBLK, NB_B), 256, smem, stream>>>(cmap, ws);
  final_kernel<<<NB_B, 256, 0, stream>>>(ws, out);
}